// Encoder_2216203125301
// MI455X (gfx1250) — compile-verified
//
#include <hip/hip_runtime.h>

#define B_    16
#define N_    512
#define DIN   256
#define DOUT  32
#define CSP   64
#define H_    256
#define W_    256
#define HW    (H_ * W_)
#define COUT  (CSP + DOUT)   // 96

typedef float v2f __attribute__((ext_vector_type(2)));
typedef float v4f __attribute__((ext_vector_type(4)));
typedef float v8f __attribute__((ext_vector_type(8)));

// ---------------------------------------------------------------------------
// Kernel 1: streaming pass over the whole output [B, 96, H, W].
// Channels [0,64): copy spatial_info. Channels [64,96): zero (scatter canvas).
// 128-bit nontemporal loads/stores; this moves ~672 MB and is the time budget.
// ---------------------------------------------------------------------------
__global__ void k_copy_zero(const v4f* __restrict__ sp, v4f* __restrict__ out) {
  size_t i = (size_t)blockIdx.x * blockDim.x + threadIdx.x;   // over B*96*HW/4
  const size_t plane4 = HW / 4;                               // 16384 (pow2)
  size_t bc  = i / plane4;
  size_t off = i - bc * plane4;
  int c = (int)(bc % COUT);
  int b = (int)(bc / COUT);
  v4f v;
  if (c < CSP) {
    v = __builtin_nontemporal_load(&sp[((size_t)b * CSP + c) * plane4 + off]);
  } else {
    v = (v4f){0.f, 0.f, 0.f, 0.f};
  }
  __builtin_nontemporal_store(v, &out[i]);
}

// ---------------------------------------------------------------------------
// Kernel 2: proj = E[8192,256] @ W[256,32] + b   via V_WMMA_F32_16X16X4_F32.
// One wave32 computes one 16x16 output tile; K consumed 4 at a time.
// A layout: lane m=lane&15 holds K = khalf, khalf+1 (contiguous 64b load).
// B layout: lane n=lane&15 holds rows K = khalf, khalf+1 of column n.
// D layout: VGPR r holds (M = r + 8*(lane>>4), N = lane&15).
// ---------------------------------------------------------------------------
__global__ void k_proj_wmma(const float* __restrict__ E,
                            const float* __restrict__ Wp,
                            const float* __restrict__ bias,
                            float* __restrict__ proj) {
  const int tilesN  = DOUT / 16;                              // 2
  int wave = blockIdx.x * (blockDim.x >> 5) + (threadIdx.x >> 5);
  int tm = wave / tilesN;                                     // 0..511
  int tn = wave - tm * tilesN;                                // 0..1
  int lane  = threadIdx.x & 31;
  int mlo   = lane & 15;
  int khalf = (lane >> 4) << 1;                               // 0 or 2
  int rowBase = tm * 16;
  int colBase = tn * 16;

  v8f acc = {};
#pragma unroll 4
  for (int k0 = 0; k0 < DIN; k0 += 4) {
    // A: 64-bit contiguous load (offset is even -> 8B aligned)
    v2f a = *reinterpret_cast<const v2f*>(E + (size_t)(rowBase + mlo) * DIN + k0 + khalf);
    // B: two rows of W, same column
    const float* bp = Wp + (size_t)(k0 + khalf) * DOUT + colBase + mlo;
    v2f b;
    b.x = bp[0];
    b.y = bp[DOUT];
    acc = __builtin_amdgcn_wmma_f32_16x16x4_f32(
        /*neg_a=*/false, a, /*neg_b=*/false, b,
        /*c_mod=*/(short)0, acc, /*reuse_a=*/false, /*reuse_b=*/false);
  }

  int mhi  = (lane >> 4) * 8;
  float bc = bias[colBase + mlo];
#pragma unroll
  for (int r = 0; r < 8; ++r) {
    int m = rowBase + mhi + r;
    proj[(size_t)m * DOUT + colBase + mlo] = acc[r] + bc;
  }
}

// ---------------------------------------------------------------------------
// Kernel 3: winner map init to -1 (n ranges 0..511, all >= 0).
// ---------------------------------------------------------------------------
__global__ void k_winner_init(int* __restrict__ winner) {
  int i = blockIdx.x * blockDim.x + threadIdx.x;
  if (i < B_ * HW) winner[i] = -1;
}

// ---------------------------------------------------------------------------
// Kernel 4: deterministic "last write wins": atomicMax over entity index n.
// ---------------------------------------------------------------------------
__global__ void k_vote(const int* __restrict__ loc, int* __restrict__ winner) {
  int bn = blockIdx.x * blockDim.x + threadIdx.x;             // 0..B*N-1
  if (bn >= B_ * N_) return;
  int b = bn / N_;
  int n = bn - b * N_;
  int h = loc[2 * bn + 0]; h = min(max(h, 0), H_ - 1);
  int w = loc[2 * bn + 1]; w = min(max(w, 0), W_ - 1);
  atomicMax(&winner[b * HW + h * W_ + w], n);
}

// ---------------------------------------------------------------------------
// Kernel 5: winners scatter their 32-channel projection into out[:, 64:96].
// 32 consecutive threads (one wave) share one entity; lane = channel.
// ---------------------------------------------------------------------------
__global__ void k_scatter(const int* __restrict__ loc, const int* __restrict__ winner,
                          const float* __restrict__ proj, float* __restrict__ out) {
  int i  = blockIdx.x * blockDim.x + threadIdx.x;             // B*N*32
  int c  = i & (DOUT - 1);
  int bn = i >> 5;
  if (bn >= B_ * N_) return;
  int b = bn / N_;
  int n = bn - b * N_;
  int h = loc[2 * bn + 0]; h = min(max(h, 0), H_ - 1);
  int w = loc[2 * bn + 1]; w = min(max(w, 0), W_ - 1);
  int flat = h * W_ + w;
  if (winner[b * HW + flat] == n) {
    out[((size_t)b * COUT + CSP + c) * HW + flat] = proj[(size_t)bn * DOUT + c];
  }
}

// ---------------------------------------------------------------------------
extern "C" void kernel_launch(void* const* d_in, const int* in_sizes, int n_in,
                              void* d_out, int out_size, void* d_ws, size_t ws_size,
                              hipStream_t stream) {
  const float* sp  = (const float*)d_in[0];   // [B, 64, H, W]
  const float* E   = (const float*)d_in[1];   // [B, N, 256]
  const float* Wp  = (const float*)d_in[2];   // [256, 32]
  const float* bp  = (const float*)d_in[3];   // [32]
  const int*   loc = (const int*)d_in[4];     // [B, N, 2]
  float* out = (float*)d_out;                 // [B, 96, H, W]

  float* proj   = (float*)d_ws;                                   // 1 MiB
  int*   winner = (int*)((char*)d_ws + (size_t)B_ * N_ * DOUT * sizeof(float)); // 4 MiB

  // 1) copy + zero the full output (bandwidth-bound main pass)
  {
    int total4 = B_ * COUT * HW / 4;                              // 25,165,824
    k_copy_zero<<<(total4 + 255) / 256, 256, 0, stream>>>((const v4f*)sp, (v4f*)out);
  }
  // 2) WMMA projection: 1024 wave-tiles, 8 waves per 256-thread block
  {
    int waves = (B_ * N_ / 16) * (DOUT / 16);                     // 1024
    k_proj_wmma<<<waves / 8, 256, 0, stream>>>(E, Wp, bp, proj);
  }
  // 3) winner init
  k_winner_init<<<(B_ * HW + 255) / 256, 256, 0, stream>>>(winner);
  // 4) vote (last-write-wins == max n, deterministic)
  k_vote<<<(B_ * N_ + 255) / 256, 256, 0, stream>>>(loc, winner);
  // 5) scatter winners into channels [64,96)
  k_scatter<<<(B_ * N_ * DOUT + 255) / 256, 256, 0, stream>>>(loc, winner, proj, out);
}